// EdgeUpdateBlock_48137993454066
// MI455X (gfx1250) — compile-verified
//
#include <hip/hip_runtime.h>
#include <hip/hip_bf16.h>

typedef __attribute__((ext_vector_type(2))) float v2f;
typedef __attribute__((ext_vector_type(4))) float v4f;
typedef __attribute__((ext_vector_type(8))) float v8f;

#define N_EDGES_C 300000
#define NTILES    18750      // N_EDGES / 16 (exact)
#define WPB       8          // waves per block (256 threads)

// D = A(16x4) * B(4x16) + C(16x16), all f32 — exact same precision as reference.
__device__ __forceinline__ v8f wmma_k4(v2f a, v2f b, v8f c) {
  return __builtin_amdgcn_wmma_f32_16x16x4_f32(false, a, false, b, (short)0, c,
                                               false, false);
}

// ---------------------------------------------------------------------------
// Pack kernel: reshuffle W_ss_ft/W_vv0_ft into per-(kstep,lane) B-fragment
// order so the main kernel's B loads are single conflict-free ds_load_b64s.
//   kappa in [0,512): kappa<256 -> ss (u=kappa>>4, v=kappa&15), else vv0.
//   Wfrag[t][ks][lane] = ( W'[4ks+2c][t*16+n], W'[4ks+2c+1][t*16+n] ),
//   c=lane>>4, n=lane&15  (matches WMMA B layout: VGPR0 holds K-rows {0|2},
//   VGPR1 K-rows {1|3}, N striped over 16 lanes).
// Tail: wsh[512] = flattened W_ss_sh / W_vv0_sh for the sh0 column.
// ---------------------------------------------------------------------------
__global__ void pack_weights(const float* __restrict__ Wss,
                             const float* __restrict__ Wvv,
                             const float* __restrict__ Wss_sh,
                             const float* __restrict__ Wvv_sh,
                             float* __restrict__ wp) {
  int i = blockIdx.x * blockDim.x + threadIdx.x;
  if (i < 16384) {                     // 2 tiles * 128 ksteps * 32 lanes * 2
    int t    = i >> 13;
    int r    = i & 8191;
    int ks   = r >> 6;
    int r2   = r & 63;
    int lane = r2 >> 1;
    int p    = r2 & 1;
    int c    = lane >> 4, n = lane & 15;
    int kk   = 4 * ks + 2 * c + p;     // kappa 0..511
    int u    = (kk >> 4) & 15;
    int v    = kk & 15;
    int w    = t * 16 + n;
    wp[i] = (kk < 256) ? Wss[u * 512 + v * 32 + w]
                       : Wvv[u * 512 + v * 32 + w];
  } else if (i < 16896) {
    int kk = i - 16384;
    int u = (kk >> 4) & 15, v = kk & 15;
    wp[i] = (kk < 256) ? Wss_sh[u * 16 + v] : Wvv_sh[u * 16 + v];
  }
}

// ---------------------------------------------------------------------------
// Main kernel: one wave32 = 16 edges. GEMM [16e x 512] x [512 x 32] via
// v_wmma_f32_16x16x4_f32; A-fragments built in registers per lane
// (lane L: edge row m=L&15, K parity pair 2c..2c+1, c=L>>4).
// sh0 accumulated alongside as a packed v2f FMA with a per-lane b64 load
// from the (L0-resident) 2 KB wsh table, then pair-reduced via shfl_xor(16).
// ---------------------------------------------------------------------------
__global__ __launch_bounds__(256) void edge_ft_kernel(
    const float* __restrict__ node_ft, const int* __restrict__ edge_index,
    const float* __restrict__ edge_ft, const float* __restrict__ edge_sh,
    const float* __restrict__ wp, float* __restrict__ out_ft,
    float* __restrict__ out_sh, const float* __restrict__ eps_ft_p,
    const float* __restrict__ eps_sh_p) {
  __shared__ v2f sB[2 * 128 * 32];     // 64 KB: B fragments [t][ks][lane]
  for (int i = threadIdx.x; i < 4096; i += 256)
    ((v4f*)sB)[i] = ((const v4f*)wp)[i];
  __syncthreads();

  const int wave = threadIdx.x >> 5, lane = threadIdx.x & 31;
  const int tile = blockIdx.x * WPB + wave;
  if (tile >= NTILES) return;          // whole-wave predication: EXEC stays all-1
  const int c = lane >> 4, n = lane & 15;

  const int e1 = tile * 16 + n;        // this lane's A-row edge
  const int snd = edge_index[e1];
  const int rcv = edge_index[N_EDGES_C + e1];
  const float* x1 = node_ft + (size_t)snd * 64;
  const float* x2 = node_ft + (size_t)rcv * 64;
  const float* wsh2 = wp + 16384 + 2 * c;   // per-lane base for sh0 weights

  v8f C0 = {}, C1 = {};
  v2f sh0v = {};
  const float inv_s3 = 0.57735026918962576f;   // 1/sqrt(3)

  // -------- ss part: ksteps 0..63, kappa = 16u + 4q + {2c,2c+1} ----------
  v2f s2p[4];
#pragma unroll
  for (int q = 0; q < 4; ++q) s2p[q] = *(const v2f*)(x2 + 4 * q + 2 * c);

  for (int u = 0; u < 16; ++u) {
    float s1u = x1[u];
#pragma unroll
    for (int q = 0; q < 4; ++q) {
      int ksg = u * 4 + q;
      v2f a;
      a.x = s1u * s2p[q].x;
      a.y = s1u * s2p[q].y;
      v2f b0 = sB[ksg * 32 + lane];
      v2f b1 = sB[4096 + ksg * 32 + lane];
      C0 = wmma_k4(a, b0, C0);
      C1 = wmma_k4(a, b1, C1);
      v2f wv = *(const v2f*)(wsh2 + 4 * ksg);
      sh0v += a * wv;
    }
  }

  // -------- vv0 part: ksteps 64..127, dot3(v1_u, v2_v)/sqrt(3) ----------
  // (1/sqrt(3) pre-folded into va/vb)
  float va[4][3], vb[4][3];
#pragma unroll
  for (int q = 0; q < 4; ++q) {
    const float* p = x2 + 16 + 3 * (4 * q + 2 * c);
#pragma unroll
    for (int j = 0; j < 3; ++j) {
      va[q][j] = p[j] * inv_s3;
      vb[q][j] = p[3 + j] * inv_s3;
    }
  }
  for (int u = 0; u < 16; ++u) {
    const float* p1 = x1 + 16 + 3 * u;
    float w0 = p1[0], w1 = p1[1], w2 = p1[2];
#pragma unroll
    for (int q = 0; q < 4; ++q) {
      int ksg = 64 + u * 4 + q;
      v2f a;
      a.x = w0 * va[q][0] + w1 * va[q][1] + w2 * va[q][2];
      a.y = w0 * vb[q][0] + w1 * vb[q][1] + w2 * vb[q][2];
      v2f b0 = sB[ksg * 32 + lane];
      v2f b1 = sB[4096 + ksg * 32 + lane];
      C0 = wmma_k4(a, b0, C0);
      C1 = wmma_k4(a, b1, C1);
      v2f wv = *(const v2f*)(wsh2 + 4 * ksg);
      sh0v += a * wv;
    }
  }

  // -------- epilogue ----------
  const float kft = 0.04419417382415922f;      // 1/(16*sqrt(2)) = norm/SQ2
  float sft = eps_ft_p[0] * kft;
  // C/D layout: VGPR g -> row M = g + 8c, col N = lane&15
#pragma unroll
  for (int g = 0; g < 8; ++g) {
    int er = tile * 16 + g + 8 * c;
    out_ft[er * 32 + n]      = edge_ft[er * 32 + n]      + sft * C0[g];
    out_ft[er * 32 + 16 + n] = edge_ft[er * 32 + 16 + n] + sft * C1[g];
  }
  float sh0p = sh0v.x + sh0v.y;
  sh0p += __shfl_xor(sh0p, 16, 32);            // combine K-parity halves
  if (c == 0) {
    int er = tile * 16 + n;
    out_sh[er * 9] = edge_sh[er * 9] + eps_sh_p[0] * kft * sh0p;
  }
}

// ---------------------------------------------------------------------------
// Tail kernel: sh1 (l=1) and sh2 (l=2), one lane per edge. Weight indices are
// lane-uniform -> scalar/constant-cache loads; per-edge matvecs in registers.
// ---------------------------------------------------------------------------
__global__ __launch_bounds__(256) void edge_sh_kernel(
    const float* __restrict__ node_ft, const int* __restrict__ edge_index,
    const float* __restrict__ edge_sh, const float* __restrict__ W_sv,
    const float* __restrict__ W_vs, const float* __restrict__ W_vv2,
    float* __restrict__ out_sh, const float* __restrict__ eps_sh_p) {
  int e = blockIdx.x * 256 + threadIdx.x;
  if (e >= N_EDGES_C) return;
  const int snd = edge_index[e], rcv = edge_index[N_EDGES_C + e];
  const float* x1 = node_ft + (size_t)snd * 64;
  const float* x2 = node_ft + (size_t)rcv * 64;

  float a1[16];
#pragma unroll
  for (int v = 0; v < 16; ++v) a1[v] = 0.f;
  for (int u = 0; u < 16; ++u) {
    float s1u = x1[u];
#pragma unroll
    for (int v = 0; v < 16; ++v) a1[v] += s1u * W_sv[u * 16 + v];
  }

  float a2[16], cm[16][3];
#pragma unroll
  for (int u = 0; u < 16; ++u) {
    a2[u] = 0.f; cm[u][0] = 0.f; cm[u][1] = 0.f; cm[u][2] = 0.f;
  }
  float sv0 = 0.f, sv1 = 0.f, sv2 = 0.f;
  for (int v = 0; v < 16; ++v) {
    float s2v = x2[v];
    float y0 = x2[16 + 3 * v], y1 = x2[17 + 3 * v], y2 = x2[18 + 3 * v];
    sv0 += a1[v] * y0; sv1 += a1[v] * y1; sv2 += a1[v] * y2;
#pragma unroll
    for (int u = 0; u < 16; ++u) {
      a2[u] += W_vs[u * 16 + v] * s2v;
      float w = W_vv2[u * 16 + v];
      cm[u][0] += w * y0; cm[u][1] += w * y1; cm[u][2] += w * y2;
    }
  }

  float vs0 = 0.f, vs1 = 0.f, vs2 = 0.f;
  float t[3][3];
#pragma unroll
  for (int i = 0; i < 3; ++i)
#pragma unroll
    for (int j = 0; j < 3; ++j) t[i][j] = 0.f;
  for (int u = 0; u < 16; ++u) {
    float z0 = x1[16 + 3 * u], z1 = x1[17 + 3 * u], z2 = x1[18 + 3 * u];
    vs0 += a2[u] * z0; vs1 += a2[u] * z1; vs2 += a2[u] * z2;
    t[0][0] += z0 * cm[u][0]; t[0][1] += z0 * cm[u][1]; t[0][2] += z0 * cm[u][2];
    t[1][0] += z1 * cm[u][0]; t[1][1] += z1 * cm[u][1]; t[1][2] += z1 * cm[u][2];
    t[2][0] += z2 * cm[u][0]; t[2][1] += z2 * cm[u][1]; t[2][2] += z2 * cm[u][2];
  }

  float eps = eps_sh_p[0];
  const float s1f = eps * 0.04419417382415922f;  // (norm/SQ2)
  const float s2f = eps * 0.0625f;               // norm
  const float rs2 = 0.7071067811865476f;         // 1/sqrt(2)
  const float rs6 = 0.4082482904638631f;         // 1/sqrt(6)
  size_t b = (size_t)e * 9;
  out_sh[b + 1] = edge_sh[b + 1] + s1f * (sv0 + vs0);
  out_sh[b + 2] = edge_sh[b + 2] + s1f * (sv1 + vs1);
  out_sh[b + 3] = edge_sh[b + 3] + s1f * (sv2 + vs2);
  // C2 (with axis map y,z,x = 0,1,2 as in reference):
  out_sh[b + 4] = edge_sh[b + 4] + s2f * rs2 * (t[2][0] + t[0][2]);
  out_sh[b + 5] = edge_sh[b + 5] + s2f * rs2 * (t[0][1] + t[1][0]);
  out_sh[b + 6] = edge_sh[b + 6] + s2f * rs6 * (2.f * t[1][1] - t[2][2] - t[0][0]);
  out_sh[b + 7] = edge_sh[b + 7] + s2f * rs2 * (t[2][1] + t[1][2]);
  out_sh[b + 8] = edge_sh[b + 8] + s2f * rs2 * (t[2][2] - t[0][0]);
}

extern "C" void kernel_launch(void* const* d_in, const int* in_sizes, int n_in,
                              void* d_out, int out_size, void* d_ws,
                              size_t ws_size, hipStream_t stream) {
  (void)in_sizes; (void)n_in; (void)out_size; (void)ws_size;
  const float* node_ft   = (const float*)d_in[0];
  const int*   edge_index= (const int*)d_in[1];
  const float* edge_ft   = (const float*)d_in[2];
  const float* edge_sh   = (const float*)d_in[3];
  const float* Wss       = (const float*)d_in[4];
  const float* Wvv       = (const float*)d_in[5];
  const float* Wss_sh    = (const float*)d_in[6];
  const float* Wvv_sh    = (const float*)d_in[7];
  const float* Wsv       = (const float*)d_in[8];
  const float* Wvs       = (const float*)d_in[9];
  const float* Wvv2      = (const float*)d_in[10];
  const float* eps_ft    = (const float*)d_in[11];
  const float* eps_sh    = (const float*)d_in[12];

  float* out_ft = (float*)d_out;
  float* out_sh = (float*)d_out + (size_t)N_EDGES_C * 32;
  float* wp     = (float*)d_ws;   // 16896 floats (66 KB) of packed weights

  pack_weights<<<(16896 + 255) / 256, 256, 0, stream>>>(Wss, Wvv, Wss_sh,
                                                        Wvv_sh, wp);
  int blocks = (NTILES + WPB - 1) / WPB;     // 2344
  edge_ft_kernel<<<blocks, 256, 0, stream>>>(node_ft, edge_index, edge_ft,
                                             edge_sh, wp, out_ft, out_sh,
                                             eps_ft, eps_sh);
  edge_sh_kernel<<<(N_EDGES_C + 255) / 256, 256, 0, stream>>>(
      node_ft, edge_index, edge_sh, Wsv, Wvs, Wvv2, out_sh, eps_sh);
}